// AdaQuadrupletMiner_65996467470341
// MI455X (gfx1250) — compile-verified
//
#include <hip/hip_runtime.h>
#include <hip/hip_bf16.h>

// ---------------------------------------------------------------------------
// AdaQuadrupletMiner for MI455X (gfx1250, wave32)
//
// Phase A (1 workgroup, 128 threads = 4 waves):
//   - row-normalize logits into LDS
//   - mat = -(x xT) via V_WMMA_F32_16X16X4_F32 (full f32 precision; the output
//     is a hard threshold vs epsilon so we keep f32 throughout)
//   - sames/diffs from labels gram (scalar, 276K MACs)
//   - epsilon via per-row prefix/suffix scans + scalar reduce
// Phase B (96x96 blocks x 256 threads): stream the 340 MB mask with
//   non-temporal 128-bit stores (output > 192 MB L2; nothing re-reads it, so
//   NT avoids L2 churn). Store-bandwidth floor ~15 us at 23.3 TB/s.
// ---------------------------------------------------------------------------

typedef float v2f __attribute__((ext_vector_type(2)));
typedef float v4f __attribute__((ext_vector_type(4)));
typedef float v8f __attribute__((ext_vector_type(8)));

#define NN 96
#define DD 64
#define CC 30

// Workspace layout (floats): [0]=epsilon, [16..] MAT (96x96), then SAME, DIFF
#define WS_MAT  16
#define WS_SAME (WS_MAT + NN*NN)
#define WS_DIFF (WS_SAME + NN*NN)

__global__ __launch_bounds__(128)
void miner_setup_kernel(const float* __restrict__ logits,
                        const float* __restrict__ labels,
                        float* __restrict__ W) {
    __shared__ float xn[NN][DD];     // 24 KB normalized rows
    __shared__ float red_s[NN];      // per-row sum1+sum2 contribution
    __shared__ float red_q[NN];      // per-row Q contribution

    float* MAT  = W + WS_MAT;
    float* SAME = W + WS_SAME;
    float* DIFF = W + WS_DIFF;

    const int tid = threadIdx.x;

    // ---- normalize rows (cosine) ----
    for (int r = tid; r < NN; r += 128) {
        float s = 0.f;
        #pragma unroll 8
        for (int k = 0; k < DD; ++k) { float v = logits[r*DD + k]; s += v*v; }
        float inv = rsqrtf(s);
        #pragma unroll 8
        for (int k = 0; k < DD; ++k) xn[r][k] = logits[r*DD + k] * inv;
    }
    __syncthreads();

    // ---- mat = -(x xT) via WMMA f32 16x16x4 ----
    // 36 tiles of 16x16; 4 waves, 9 tiles each; K=64 -> 16 wmma steps/tile.
    {
        const int wave = tid >> 5;
        const int lane = tid & 31;
        const int half = lane >> 4;      // 0: K pair {0,1}; 1: K pair {2,3}
        const int lm   = lane & 15;
        for (int t = wave; t < 36; t += 4) {     // wave-uniform: EXEC all ones
            const int tm = t / 6, tn = t % 6;
            const int m = tm*16 + lm;            // A row for this lane
            const int n = tn*16 + lm;            // B col for this lane
            v8f acc = {};
            #pragma unroll
            for (int kb = 0; kb < DD; kb += 4) {
                v2f a, b;
                a.x = xn[m][kb + 2*half];
                a.y = xn[m][kb + 2*half + 1];
                b.x = xn[n][kb + 2*half];        // B[k][n] = xn[n][k]
                b.y = xn[n][kb + 2*half + 1];
                acc = __builtin_amdgcn_wmma_f32_16x16x4_f32(
                        /*neg_a=*/false, a, /*neg_b=*/false, b,
                        /*c_mod=*/(short)0, acc,
                        /*reuse_a=*/false, /*reuse_b=*/false);
            }
            #pragma unroll
            for (int r = 0; r < 8; ++r) {
                const int M = tm*16 + r + 8*half;      // C/D layout: VGPR r
                MAT[M*NN + tn*16 + lm] = -acc[r];      // mat = -(x xT)
            }
        }
    }
    __syncthreads();

    // ---- sames_raw = (labels labelsT) > 0 ; diffs = !sames_raw ;
    //      sames = sames_raw with zero diagonal ----
    for (int p = tid; p < NN*NN; p += 128) {
        const int i = p / NN, j = p % NN;
        float dot = 0.f;
        for (int c = 0; c < CC; ++c) dot += labels[i*CC + c] * labels[j*CC + c];
        const float sraw = (dot > 0.f) ? 1.f : 0.f;
        DIFF[p] = 1.f - sraw;
        SAME[p] = (i == j) ? 0.f : sraw;
    }
    __syncthreads();

    // ---- epsilon: per-row prefix/suffix scans, then scalar reduce ----
    if (tid < NN) {
        const int i = tid;
        const float* sf = SAME + i*NN;
        const float* df = DIFF + i*NN;
        const float* mr = MAT  + i*NN;
        // cnt_k[j] = sum_{k>j} sf[k]  (suffix)
        float suf = 0.f, w1sum = 0.f, mw1 = 0.f;
        for (int j = NN-1; j >= 0; --j) {
            w1sum += sf[j] * suf;
            mw1   += mr[j] * sf[j] * suf;
            suf   += sf[j];
        }
        // cnt_j[k] = sum_{j<k} sf[j]  (prefix)
        float pre = 0.f, w2sum = 0.f, mw2 = 0.f;
        for (int k = 0; k < NN; ++k) {
            w2sum += sf[k] * pre;
            mw2   += mr[k] * sf[k] * pre;
            pre   += sf[k];
        }
        float ds = 0.f, mds = 0.f;
        for (int n = 0; n < NN; ++n) { ds += df[n]; mds += mr[n] * df[n]; }
        red_s[i] = (mds * w1sum - mw1 * ds) + (mds * w2sum - mw2 * ds);
        red_q[i] = w1sum * ds;
    }
    __syncthreads();
    if (tid == 0) {
        float s12 = 0.f, Q = 0.f;
        for (int i = 0; i < NN; ++i) { s12 += red_s[i]; Q += red_q[i]; }
        const float mean_delta = s12 / fmaxf(2.f * Q, 1.f);
        W[0] = fmaxf(mean_delta / 2.0f, 0.f);     // relu(mean_delta / K_DELTA)
    }
}

// One block per (i, j). mask[i,j,k,n] = c[i,j,n] & c[i,k,n] & (k > j)
// where c[i,p,n] = sames[i,p] & diffs[i,n] & (0 < mat[i,n]-mat[i,p] <= eps).
__global__ __launch_bounds__(256)
void miner_mask_kernel(const float* __restrict__ W,
                       float* __restrict__ out) {
    __shared__ float matrow[NN];
    __shared__ float srow[NN];
    __shared__ float c1[NN];     // c[i, j, n] for this block's (i, j)

    const float* MAT  = W + WS_MAT;
    const float* SAME = W + WS_SAME;
    const float* DIFF = W + WS_DIFF;

    const int j = blockIdx.x;
    const int i = blockIdx.y;
    const int tid = threadIdx.x;
    const float eps = W[0];

    if (tid < NN) {
        matrow[tid] = MAT[i*NN + tid];
        srow[tid]   = SAME[i*NN + tid];
    }
    __syncthreads();

    const float sij = srow[j];
    const float mj  = matrow[j];
    if (tid < NN) {
        const float m = matrow[tid] - mj;
        c1[tid] = (sij != 0.f && DIFF[i*NN + tid] != 0.f &&
                   m > 0.f && m <= eps) ? 1.f : 0.f;
    }
    __syncthreads();

    // 96 (k) x 24 (n in v4f) = 2304 coalesced non-temporal 128-bit stores
    v4f* outp = (v4f*)(out + (size_t)(i*NN + j) * (NN*NN));
    for (int idx = tid; idx < NN*(NN/4); idx += 256) {
        const int k  = idx / (NN/4);
        const int n4 = idx % (NN/4);
        const float sk = (k > j) ? srow[k] : 0.f;
        const float mk = matrow[k];
        v4f v;
        #pragma unroll
        for (int q = 0; q < 4; ++q) {
            const int n = n4*4 + q;
            const float m = matrow[n] - mk;
            v[q] = (sk != 0.f && c1[n] != 0.f && m > 0.f && m <= eps) ? 1.f : 0.f;
        }
        __builtin_nontemporal_store(v, &outp[idx]);   // global_store_b128 TH_NT
    }
}

extern "C" void kernel_launch(void* const* d_in, const int* in_sizes, int n_in,
                              void* d_out, int out_size, void* d_ws, size_t ws_size,
                              hipStream_t stream) {
    const float* logits = (const float*)d_in[0];   // [96, 64] f32
    const float* labels = (const float*)d_in[1];   // [96, 30] f32
    float* out = (float*)d_out;                    // [96,96,96,96] f32
    float* W   = (float*)d_ws;                     // >= (16 + 3*9216) floats

    miner_setup_kernel<<<dim3(1), dim3(128), 0, stream>>>(logits, labels, W);
    miner_mask_kernel<<<dim3(NN, NN), dim3(256), 0, stream>>>(W, out);
}